// LBPPooling_26654567039297
// MI455X (gfx1250) — compile-verified
//
#include <hip/hip_runtime.h>
#include <stdint.h>

// ---------------------------------------------------------------------------
// LBP + per-channel BN + 2x2 avgpool for x=(32,64,128,128) f32.
// Pass1: per-plane LBP codes + sum/sumsq partials (deterministic reduce),
//        optionally spilling packed u8 codes to workspace.
// Pass2: per-channel scale/shift.
// Pass3: pool from stored codes (fast path) or recompute (fallback).
// CDNA5 path: GLOBAL_LOAD_ASYNC_TO_LDS_B128 staging of each plane into LDS.
// ---------------------------------------------------------------------------

#define BATCH 32
#define CCH   64
#define HH    128
#define WW    128
#define PLANE (HH * WW)
#define NPERC (BATCH * HH * WW)      // 524288 values per channel
#define LS    136                    // LDS tile row stride (floats)
#define LROWS 130                    // 128 rows + top/bottom halo
#define COFF  4                      // interior column offset (keeps b128 dst 16B aligned)

// ws layout (floats): [0,2048) sums, [2048,4096) sumsqs, [4096,4224) scale/shift,
// then (optional) packed LBP codes: 2048 planes * 4096 u32 = 32 MiB.
#define WS_CODES_F32_OFF 4224
#define WS_NEED_BYTES ((size_t)WS_CODES_F32_OFF * 4 + (size_t)2048 * 4096 * 4)

typedef int vi4 __attribute__((vector_size(16)));
typedef __attribute__((address_space(1))) vi4 as1_vi4;
typedef __attribute__((address_space(3))) vi4 as3_vi4;

#if __has_builtin(__builtin_amdgcn_global_load_async_to_lds_b128)
#define ASYNC_LDS_B128(gsrc, ldst)                                            \
  __builtin_amdgcn_global_load_async_to_lds_b128(                             \
      (as1_vi4*)(void*)(gsrc), (as3_vi4*)(void*)(ldst), 0, 0)
#else
#define ASYNC_LDS_B128(gsrc, ldst)                                            \
  (*(float4*)(ldst) = *(const float4*)(gsrc))
#endif

#if __has_builtin(__builtin_amdgcn_s_wait_asynccnt)
#define ASYNC_WAIT() __builtin_amdgcn_s_wait_asynccnt(0)
#elif __has_builtin(__builtin_amdgcn_global_load_async_to_lds_b128)
#define ASYNC_WAIT() asm volatile("s_wait_asynccnt 0" ::: "memory")
#else
#define ASYNC_WAIT() ((void)0)
#endif

// Stage one 128x128 plane into LDS with a zeroed 1-element halo.
__device__ __forceinline__ void load_plane_tile(const float* __restrict__ src,
                                                float* tile, int tid) {
  for (int i = tid; i < 2 * LS; i += 256) {          // top & bottom halo rows
    int r = (i < LS) ? 0 : (LROWS - 1);
    int c = (i < LS) ? i : (i - LS);
    tile[r * LS + c] = 0.0f;
  }
  {                                                  // left/right halo columns
    int r = (tid & 127) + 1;
    int c = (tid < 128) ? (COFF - 1) : (COFF + WW);
    tile[r * LS + c] = 0.0f;
  }
  for (int it = 0; it < 16; ++it) {                  // async 128 rows x 512B
    int task = it * 256 + tid;
    int row = task >> 5;
    int seg = task & 31;
    const float* g = src + row * WW + seg * 4;
    float* l = tile + (row + 1) * LS + COFF + seg * 4;  // 16B-aligned dst
    ASYNC_LDS_B128(g, l);
  }
  ASYNC_WAIT();
  __syncthreads();
}

struct Row6 { float l, m0, m1, m2, m3, r; };

__device__ __forceinline__ Row6 load_row6(const float* tp) {
  Row6 v;
  v.l = tp[0];
  float4 q = *(const float4*)(tp + 1);   // ds_load_b128, 16B aligned
  v.m0 = q.x; v.m1 = q.y; v.m2 = q.z; v.m3 = q.w;
  v.r = tp[5];
  return v;
}

// Neighbor weights: (-1,-1)=1 (-1,0)=2 (-1,1)=4 (0,-1)=8 (0,1)=16
// (1,-1)=32 (1,0)=64 (1,1)=128, predicate nb >= center.
__device__ __forceinline__ unsigned lbp8(float aL, float aC, float aR,
                                         float bL, float ct, float bR,
                                         float cL, float cC, float cR) {
  unsigned code = 0u;
  code |= (aL >= ct) ? 1u   : 0u;
  code |= (aC >= ct) ? 2u   : 0u;
  code |= (aR >= ct) ? 4u   : 0u;
  code |= (bL >= ct) ? 8u   : 0u;
  code |= (bR >= ct) ? 16u  : 0u;
  code |= (cL >= ct) ? 32u  : 0u;
  code |= (cC >= ct) ? 64u  : 0u;
  code |= (cR >= ct) ? 128u : 0u;
  return code;
}

#define CODES4(A, Bm, Cm, c0, c1, c2, c3)                                     \
  c0 = lbp8(A.l,  A.m0, A.m1, Bm.l,  Bm.m0, Bm.m1, Cm.l,  Cm.m0, Cm.m1);      \
  c1 = lbp8(A.m0, A.m1, A.m2, Bm.m0, Bm.m1, Bm.m2, Cm.m0, Cm.m1, Cm.m2);      \
  c2 = lbp8(A.m1, A.m2, A.m3, Bm.m1, Bm.m2, Bm.m3, Cm.m1, Cm.m2, Cm.m3);      \
  c3 = lbp8(A.m2, A.m3, A.r,  Bm.m2, Bm.m3, Bm.r,  Cm.m2, Cm.m3, Cm.r)

// ---------------------------------------------------------------------------
// Pass 1: per-plane sum/sumsq of lbp/255; optionally store packed codes.
// ---------------------------------------------------------------------------
template <bool STORE_CODES>
__global__ __launch_bounds__(256) void lbp_stats_kernel(
    const float* __restrict__ x, float* __restrict__ partial,
    uint32_t* __restrict__ codes) {
  __shared__ float tile[LROWS * LS];
  __shared__ float red[512];
  const int p = blockIdx.x;                    // plane = b*64 + c
  const int tid = threadIdx.x;
  const float* src = x + (size_t)p * PLANE;
  load_plane_tile(src, tile, tid);

  const int cg = tid & 31;                     // 4-column group
  const int rg = tid >> 5;                     // 16-row group
  const int r0 = rg * 16;
  const float* base = tile + (COFF - 1) + cg * 4;
  uint32_t* cptr = STORE_CODES ? (codes + (size_t)p * 4096 + r0 * 32 + cg) : nullptr;

  const float inv255 = 1.0f / 255.0f;
  float sum = 0.0f, sumsq = 0.0f;
  Row6 A  = load_row6(base + (r0 + 0) * LS);
  Row6 Bm = load_row6(base + (r0 + 1) * LS);
#pragma unroll
  for (int rr = 0; rr < 16; ++rr) {
    Row6 Cm = load_row6(base + (r0 + rr + 2) * LS);
    unsigned c0, c1, c2, c3;
    CODES4(A, Bm, Cm, c0, c1, c2, c3);
    if (STORE_CODES) cptr[rr * 32] = c0 | (c1 << 8) | (c2 << 16) | (c3 << 24);
    float f0 = (float)c0 * inv255, f1 = (float)c1 * inv255;
    float f2 = (float)c2 * inv255, f3 = (float)c3 * inv255;
    sum += (f0 + f1) + (f2 + f3);
    sumsq = fmaf(f0, f0, sumsq); sumsq = fmaf(f1, f1, sumsq);
    sumsq = fmaf(f2, f2, sumsq); sumsq = fmaf(f3, f3, sumsq);
    A = Bm; Bm = Cm;
  }

  red[tid] = sum;
  red[256 + tid] = sumsq;
  __syncthreads();
  for (int s = 128; s > 0; s >>= 1) {
    if (tid < s) {
      red[tid] += red[tid + s];
      red[256 + tid] += red[256 + tid + s];
    }
    __syncthreads();
  }
  if (tid == 0) {
    const int c = p & 63, b = p >> 6;
    partial[c * 32 + b] = red[0];
    partial[2048 + c * 32 + b] = red[256];
  }
}

// ---------------------------------------------------------------------------
// Pass 2: fold 32 partials per channel (fixed order) -> scale/shift
// ---------------------------------------------------------------------------
__global__ __launch_bounds__(64) void stats_finalize_kernel(
    const float* __restrict__ partial, const float* __restrict__ gamma,
    const float* __restrict__ beta, float* __restrict__ scale_shift) {
  const int c = threadIdx.x;
  float s = 0.0f, ss = 0.0f;
  for (int b = 0; b < 32; ++b) {
    s += partial[c * 32 + b];
    ss += partial[2048 + c * 32 + b];
  }
  const float invN = 1.0f / (float)NPERC;
  const float mean = s * invN;
  const float var = fmaf(-mean, mean, ss * invN);
  const float sc = gamma[c] * rsqrtf(var + 1e-5f);
  const float sh = fmaf(-mean, sc, beta[c]);
  scale_shift[c] = sc;
  scale_shift[64 + c] = sh;
}

// ---------------------------------------------------------------------------
// Pass 3 (fast): pool directly from stored codes.
// out(or,oc) = (sum of 4 codes) * (scale/1020) + shift
// ---------------------------------------------------------------------------
__device__ __forceinline__ void pair_from_words(uint32_t wt, uint32_t wb,
                                                float k, float sh,
                                                float& a, float& b) {
  uint32_t s0 = (wt & 0xFFu) + ((wt >> 8) & 0xFFu) +
                (wb & 0xFFu) + ((wb >> 8) & 0xFFu);
  uint32_t s1 = ((wt >> 16) & 0xFFu) + (wt >> 24) +
                ((wb >> 16) & 0xFFu) + (wb >> 24);
  a = fmaf((float)s0, k, sh);
  b = fmaf((float)s1, k, sh);
}

__global__ __launch_bounds__(256) void lbp_pool_codes_kernel(
    const uint32_t* __restrict__ codes, const float* __restrict__ scale_shift,
    float* __restrict__ out) {
  const int t = blockIdx.x * 256 + threadIdx.x;  // 0 .. 1048575 (8 outputs each)
  const int p = t >> 9;                          // plane
  const int tp = t & 511;
  const int orow = tp >> 3;                      // output row 0..63
  const int q = tp & 7;                          // output col group (8 cols)
  const int ch = p & 63;
  const float sc = scale_shift[ch];
  const float sh = scale_shift[64 + ch];
  const float k = sc * (1.0f / (255.0f * 4.0f));

  // code row 2*orow starts at u32 index orow*64 within the plane
  const uint32_t* cb = codes + (size_t)p * 4096 + orow * 64 + q * 4;
  uint4 topq = *(const uint4*)(cb);        // 16 codes, rows 2*orow
  uint4 botq = *(const uint4*)(cb + 32);   // 16 codes, rows 2*orow+1

  float4 oA, oB;
  pair_from_words(topq.x, botq.x, k, sh, oA.x, oA.y);
  pair_from_words(topq.y, botq.y, k, sh, oA.z, oA.w);
  pair_from_words(topq.z, botq.z, k, sh, oB.x, oB.y);
  pair_from_words(topq.w, botq.w, k, sh, oB.z, oB.w);

  float* op = out + (size_t)p * 4096 + orow * 64 + q * 8;
  *(float4*)op = oA;
  *(float4*)(op + 4) = oB;
}

// ---------------------------------------------------------------------------
// Pass 3 (fallback): recompute codes from x and pool.
// ---------------------------------------------------------------------------
__global__ __launch_bounds__(256) void lbp_pool_kernel(
    const float* __restrict__ x, const float* __restrict__ scale_shift,
    float* __restrict__ out) {
  __shared__ float tile[LROWS * LS];
  const int p = blockIdx.x;
  const int tid = threadIdx.x;
  const int ch = p & 63;
  const float* src = x + (size_t)p * PLANE;
  load_plane_tile(src, tile, tid);

  const float sc = scale_shift[ch];
  const float sh = scale_shift[64 + ch];
  const float k = sc * (1.0f / (255.0f * 4.0f));

  const int cg = tid & 31;
  const int rg = tid >> 5;
  const int r0 = rg * 16;
  const float* base = tile + (COFF - 1) + cg * 4;
  float* outp = out + (size_t)p * 4096 + (size_t)(r0 >> 1) * 64 + cg * 2;

  Row6 A  = load_row6(base + (r0 + 0) * LS);
  Row6 Bm = load_row6(base + (r0 + 1) * LS);
  unsigned t0 = 0, t1 = 0;
#pragma unroll
  for (int rr = 0; rr < 16; ++rr) {
    Row6 Cm = load_row6(base + (r0 + rr + 2) * LS);
    unsigned c0, c1, c2, c3;
    CODES4(A, Bm, Cm, c0, c1, c2, c3);
    if ((rr & 1) == 0) {
      t0 = c0 + c1;
      t1 = c2 + c3;
    } else {
      t0 += c0 + c1;
      t1 += c2 + c3;
      float2 o;
      o.x = fmaf((float)t0, k, sh);
      o.y = fmaf((float)t1, k, sh);
      *(float2*)(outp + (rr >> 1) * 64) = o;
    }
    A = Bm; Bm = Cm;
  }
}

// ---------------------------------------------------------------------------
// Host launch.
// ---------------------------------------------------------------------------
extern "C" void kernel_launch(void* const* d_in, const int* in_sizes, int n_in,
                              void* d_out, int out_size, void* d_ws,
                              size_t ws_size, hipStream_t stream) {
  const float* x = (const float*)d_in[0];
  const float* gamma = (const float*)d_in[1];
  const float* beta = (const float*)d_in[2];
  float* out = (float*)d_out;
  float* ws = (float*)d_ws;
  uint32_t* codes = (uint32_t*)(ws + WS_CODES_F32_OFF);

  const bool use_codes = (ws_size >= WS_NEED_BYTES);

  if (use_codes) {
    lbp_stats_kernel<true><<<BATCH * CCH, 256, 0, stream>>>(x, ws, codes);
    stats_finalize_kernel<<<1, 64, 0, stream>>>(ws, gamma, beta, ws + 4096);
    lbp_pool_codes_kernel<<<4096, 256, 0, stream>>>(codes, ws + 4096, out);
  } else {
    lbp_stats_kernel<false><<<BATCH * CCH, 256, 0, stream>>>(x, ws, nullptr);
    stats_finalize_kernel<<<1, 64, 0, stream>>>(ws, gamma, beta, ws + 4096);
    lbp_pool_kernel<<<BATCH * CCH, 256, 0, stream>>>(x, ws + 4096, out);
  }
}